// spatial_attention_13125420056715
// MI455X (gfx1250) — compile-verified
//
#include <hip/hip_runtime.h>

// Problem constants (match reference)
#define BB   8
#define CC   64
#define NN   2000
#define TT   12
#define EE   64000
#define HID  32
#define NT16 (NN / 16)        // 125 tiles of 16 nodes

typedef __attribute__((ext_vector_type(2))) float v2f;
typedef __attribute__((ext_vector_type(4))) float v4f;
typedef __attribute__((ext_vector_type(8))) float v8f;

// ---------------------------------------------------------------------------
// K1: fold W3/W4/conv_w/scales into two 64x64 matrices
//   M1[c][c'] = W1s * sum_h W3[c][h]   * conv_w[c'][h]
//   M2[c][c'] = W2s * sum_h W4[c][h]   * conv_w[c'][32+h]
// ---------------------------------------------------------------------------
__global__ __launch_bounds__(256) void k1_fold(
    const float* __restrict__ W3, const float* __restrict__ W4,
    const float* __restrict__ conv_w,
    const float* __restrict__ W1s, const float* __restrict__ W2s,
    float* __restrict__ M1, float* __restrict__ M2)
{
    int t   = blockIdx.x * 256 + threadIdx.x;   // 0..8191
    int m   = t >> 12;                          // 0 -> M1, 1 -> M2
    int rem = t & 4095;
    int c   = rem >> 6;
    int cp  = rem & 63;
    const float* wa = (m == 0 ? W3 : W4) + c * HID;
    const float* wb = conv_w + cp * (2 * HID) + (m == 0 ? 0 : HID);
    float s = 0.0f;
#pragma unroll
    for (int h = 0; h < HID; ++h) s += wa[h] * wb[h];
    float scale = (m == 0 ? W1s[0] : W2s[0]);
    (m == 0 ? M1 : M2)[c * 64 + cp] = scale * s;
}

// ---------------------------------------------------------------------------
// K2: one wave32 per (batch b, 16-node tile).
//   up[n][c] = sum_t x[b][c][n][t] * (t/11)
//   dn[n][c] = sum_t x[b][c][n][t] * (1 - t/11)  = (sum_t x) - up
//   P[b][n][:] = up @ M1 ;  Q[b][n][:] = dn @ M2  via V_WMMA_F32_16X16X4_F32
// A-frag layout (32-bit A 16x4): lane = M (mod 16), VGPR v holds K = v + 2*(lane/16)
// B-frag layout (4x16):          VGPR v holds row K = v + 2*(lane/16), lane%16 = N
// C/D (16x16 f32): VGPR r -> M = r + 8*(lane/16), lane%16 = N
// x is streamed exactly once -> non-temporal loads so it never displaces
// M1/M2/P/Q from L2 (needed hot for the gather phase).
// ---------------------------------------------------------------------------
__global__ __launch_bounds__(32) void k2_time_gemm(
    const float* __restrict__ x,
    const float* __restrict__ M1g, const float* __restrict__ M2g,
    float* __restrict__ P, float* __restrict__ Q)
{
    const int b    = blockIdx.x / NT16;
    const int n0   = (blockIdx.x % NT16) * 16;
    const int l    = threadIdx.x;
    const int M    = l & 15;
    const int half = l >> 4;

    const v8f vzero = {0.f, 0.f, 0.f, 0.f, 0.f, 0.f, 0.f, 0.f};
    v8f accP[4], accQ[4];
#pragma unroll
    for (int j = 0; j < 4; ++j) { accP[j] = vzero; accQ[j] = vzero; }

    const float inv11 = 1.0f / 11.0f;

#pragma unroll 1
    for (int kk = 0; kk < 16; ++kk) {           // K = 64 channels, 4 per step
        v2f a_up, a_dn;
#pragma unroll
        for (int v = 0; v < 2; ++v) {
            const int c = kk * 4 + v + 2 * half;
            const v4f* px = (const v4f*)(x + (size_t)(((b * CC + c) * NN + (n0 + M)) * TT));
            v4f x0 = __builtin_nontemporal_load(px + 0);
            v4f x1 = __builtin_nontemporal_load(px + 1);
            v4f x2 = __builtin_nontemporal_load(px + 2);
            float s  = x0[0] + x0[1] + x0[2] + x0[3]
                     + x1[0] + x1[1] + x1[2] + x1[3]
                     + x2[0] + x2[1] + x2[2] + x2[3];
            float su =              x0[1] * 1.f + x0[2] * 2.f  + x0[3] * 3.f
                     + x1[0] * 4.f + x1[1] * 5.f + x1[2] * 6.f  + x1[3] * 7.f
                     + x2[0] * 8.f + x2[1] * 9.f + x2[2] * 10.f + x2[3] * 11.f;
            su *= inv11;
            a_up[v] = su;        // weights t/11
            a_dn[v] = s - su;    // weights (11-t)/11
        }
        const int r0 = (kk * 4 + 2 * half) * 64;   // B-frag row K for VGPR 0
        const int r1 = r0 + 64;                    // row K+1 for VGPR 1
#pragma unroll
        for (int j = 0; j < 4; ++j) {              // 4 tiles of 16 output cols
            const int col = j * 16 + M;
            v2f b1, b2;
            b1[0] = M1g[r0 + col];
            b1[1] = M1g[r1 + col];
            b2[0] = M2g[r0 + col];
            b2[1] = M2g[r1 + col];
            accP[j] = __builtin_amdgcn_wmma_f32_16x16x4_f32(
                false, a_up, false, b1, (short)0, accP[j], false, false);
            accQ[j] = __builtin_amdgcn_wmma_f32_16x16x4_f32(
                false, a_dn, false, b2, (short)0, accQ[j], false, false);
        }
    }

    // Store 16x64 tiles of P and Q (regular temporal: keep resident in L2)
#pragma unroll
    for (int j = 0; j < 4; ++j) {
        const int col = j * 16 + M;
#pragma unroll
        for (int r = 0; r < 8; ++r) {
            const int row = n0 + r + 8 * half;
            const size_t o = (size_t)(b * NN + row) * 64 + col;
            P[o] = accP[j][r];
            Q[o] = accQ[j][r];
        }
    }
}

// ---------------------------------------------------------------------------
// K3: out[b][e][:] = P[b][idx[e]][:] + Q[b][idy[e]][:] + conv_b
// 16 lanes (float4 each) per output row; 16 rows per 256-thread block.
// P/Q gathers hit L2 (8 MB resident); out (131 MB, write-once) is stored
// non-temporally so it does not evict the gather sources from L2.
// ---------------------------------------------------------------------------
__global__ __launch_bounds__(256) void k3_gather(
    const float* __restrict__ P, const float* __restrict__ Q,
    const int* __restrict__ idx, const int* __restrict__ idy,
    const float* __restrict__ conv_b, float* __restrict__ out)
{
    const int rid  = blockIdx.x * 16 + (threadIdx.x >> 4);  // row in [0, B*E)
    const int lane = threadIdx.x & 15;
    const int b = rid / EE;
    const int e = rid - b * EE;
    const int i1 = idx[e];
    const int i2 = idy[e];

    const v4f* p4  = (const v4f*)(P + (size_t)(b * NN + i1) * 64) + lane;
    const v4f* q4  = (const v4f*)(Q + (size_t)(b * NN + i2) * 64) + lane;
    const v4f* cb4 = (const v4f*)conv_b + lane;
    v4f p = *p4, q = *q4, cb = *cb4;
    v4f o = p + q + cb;
    __builtin_nontemporal_store(o, (v4f*)out + (size_t)rid * 16 + lane);
}

// ---------------------------------------------------------------------------
extern "C" void kernel_launch(void* const* d_in, const int* in_sizes, int n_in,
                              void* d_out, int out_size, void* d_ws, size_t ws_size,
                              hipStream_t stream)
{
    const float* x      = (const float*)d_in[0];
    const int*   idx    = (const int*)  d_in[1];
    const int*   idy    = (const int*)  d_in[2];
    const float* W1s    = (const float*)d_in[3];
    const float* W2s    = (const float*)d_in[4];
    const float* W3     = (const float*)d_in[5];
    const float* W4     = (const float*)d_in[6];
    const float* conv_w = (const float*)d_in[7];
    const float* conv_b = (const float*)d_in[8];
    float* out = (float*)d_out;

    // workspace layout: M1 (16KB) | M2 (16KB) | P (4MB) | Q (4MB)
    float* M1 = (float*)d_ws;
    float* M2 = M1 + 64 * 64;
    float* P  = M2 + 64 * 64;
    float* Q  = P + (size_t)BB * NN * 64;

    k1_fold<<<32, 256, 0, stream>>>(W3, W4, conv_w, W1s, W2s, M1, M2);
    k2_time_gemm<<<BB * NT16, 32, 0, stream>>>(x, M1, M2, P, Q);
    k3_gather<<<(BB * EE) / 16, 256, 0, stream>>>(P, Q, idx, idy, conv_b, out);
}